// VectorQuantizer_25855703122382
// MI455X (gfx1250) — compile-verified
//
#include <hip/hip_runtime.h>
#include <math.h>

typedef __attribute__((ext_vector_type(2))) float v2f;
typedef __attribute__((ext_vector_type(8))) float v8f;

#define NE     16384      // codebook entries
#define EDIM   256        // embedding dim
#define BATCH  4
#define NTOK   2048       // tokens per batch item
#define TTOT   8192       // BATCH*NTOK
#define LDA    260        // padded LDS row stride (floats) for A tile: 260%4==0 (float4 ok),
                          // 260 mod 64 = 4 -> fragment loads hit distinct banks

// d_out layout (float elements), reference tuple order:
// quant[4,256,2048], vq_loss, commit_loss, usage, perplexity,
// min_encodings[8192,16384], encoding_indices[8192]
static constexpr size_t OFF_QUANT  = 0;
static constexpr size_t OFF_VQ     = 2097152;
static constexpr size_t OFF_COMMIT = 2097153;
static constexpr size_t OFF_USAGE  = 2097154;
static constexpr size_t OFF_PERP   = 2097155;
static constexpr size_t OFF_MINENC = 2097156;
static constexpr size_t OFF_IDX    = OFF_MINENC + (size_t)TTOT * NE; // 136314884

// ---------------------------------------------------------------------------
// 1) Token L2-normalize: z[b, c, n] -> zn[t, c], t = b*2048 + n. Wave per token.
// ---------------------------------------------------------------------------
__global__ __launch_bounds__(256)
void k_norm_z(const float* __restrict__ z, float* __restrict__ zn) {
  const int t    = blockIdx.x * 8 + (threadIdx.x >> 5);
  const int lane = threadIdx.x & 31;
  const int b = t >> 11;
  const int n = t & 2047;
  const float* src = z + (size_t)b * EDIM * NTOK + n;
  float vals[8];
  float ss = 0.0f;
#pragma unroll
  for (int i = 0; i < 8; ++i) {
    vals[i] = src[(size_t)(lane + 32 * i) * NTOK];
    ss += vals[i] * vals[i];
  }
#pragma unroll
  for (int m = 1; m < 32; m <<= 1) ss += __shfl_xor(ss, m, 32);
  const float inv = 1.0f / fmaxf(sqrtf(ss), 1e-12f);
  float* dst = zn + (size_t)t * EDIM;
#pragma unroll
  for (int i = 0; i < 8; ++i) dst[lane + 32 * i] = vals[i] * inv;
}

// ---------------------------------------------------------------------------
// 2a) Codebook row inverse norms. Wave per code row.
// ---------------------------------------------------------------------------
__global__ __launch_bounds__(256)
void k_emb_norms(const float* __restrict__ emb, float* __restrict__ invn) {
  const int j    = blockIdx.x * 8 + (threadIdx.x >> 5);
  const int lane = threadIdx.x & 31;
  const float* src = emb + (size_t)j * EDIM;
  float ss = 0.0f;
#pragma unroll
  for (int i = 0; i < 8; ++i) { const float v = src[lane + 32 * i]; ss += v * v; }
#pragma unroll
  for (int m = 1; m < 32; m <<= 1) ss += __shfl_xor(ss, m, 32);
  if (lane == 0) invn[j] = 1.0f / fmaxf(sqrtf(ss), 1e-12f);
}

// ---------------------------------------------------------------------------
// 2b) Normalized transpose: embT[c, j] = emb[j, c] * invn[j]. 32x32 LDS tiles.
// ---------------------------------------------------------------------------
__global__ __launch_bounds__(256)
void k_transpose(const float* __restrict__ emb, const float* __restrict__ invn,
                 float* __restrict__ embT) {
  __shared__ float tile[32][33];
  const int j0 = blockIdx.x * 32;
  const int c0 = blockIdx.y * 32;
  const int jl = threadIdx.x >> 5;   // 0..7
  const int cl = threadIdx.x & 31;
#pragma unroll
  for (int ii = 0; ii < 4; ++ii) {
    const int j = jl + ii * 8;
    tile[j][cl] = emb[(size_t)(j0 + j) * EDIM + c0 + cl];
  }
  __syncthreads();
  const int tj = threadIdx.x & 31;   // code index within tile (coalesced writes)
  const int tc = threadIdx.x >> 5;
  const float s = invn[j0 + tj];
#pragma unroll
  for (int ii = 0; ii < 4; ++ii) {
    const int c = tc + ii * 8;
    embT[(size_t)(c0 + c) * NE + j0 + tj] = tile[tj][c] * s;
  }
}

// ---------------------------------------------------------------------------
// 3) f32 WMMA GEMM + running argmax.
//    Grid: (TTOT/64, 4). Block: 256 thr = 8 waves. Block owns 64 tokens and
//    codes [y*4096, (y+1)*4096). Each wave owns a 16-code strip per 128-code
//    chunk and computes four 16x16 tiles (64 tokens) via V_WMMA_F32_16X16X4_F32,
//    K-stepped by 4 over EDIM=256.
//    Fragment layouts per CDNA5 ISA 7.12.2:
//      A 16x4: lane m(0..15) v0=A[m][0] v1=A[m][1]; lane m+16: K=2,3
//      B 4x16: lane n(0..15) v0=B[0][n] v1=B[1][n]; lane n+16: K=2,3
//      C/D:    vgpr r, lanes0-15 -> M=r, lanes16-31 -> M=8+r, N=lane%16
// ---------------------------------------------------------------------------
__global__ __launch_bounds__(256)
void k_gemm_argmax(const float* __restrict__ zn, const float* __restrict__ embT,
                   float* __restrict__ pbv, int* __restrict__ pbi) {
  __shared__ float As[64 * LDA];
  __shared__ float wbv[8 * 64];
  __shared__ int   wbi[8 * 64];

  const int tid  = threadIdx.x;
  const int w    = tid >> 5;
  const int lane = tid & 31;
  const int h    = lane >> 4;   // K-half selector
  const int nn   = lane & 15;   // column / row-within-fragment
  const int t0   = blockIdx.x * 64;
  const int yb   = blockIdx.y * 4096;

  // Cooperative A-tile load: 64 tokens x 256 floats (float4, coalesced).
  for (int it = 0; it < 16; ++it) {
    const int elem = it * 256 + tid;
    const int row  = elem >> 6;
    const int col  = (elem & 63) << 2;
    const float4 v = *(const float4*)(zn + (size_t)(t0 + row) * EDIM + col);
    *(float4*)(&As[row * LDA + col]) = v;
  }
  __syncthreads();

  float bv[32];
  int   bi[32];
#pragma unroll
  for (int q = 0; q < 32; ++q) { bv[q] = -3.0e38f; bi[q] = 0x7fffffff; }

  for (int chunk = 0; chunk < 32; ++chunk) {
    const int j = yb + chunk * 128 + w * 16 + nn;      // this lane's code column
    const float* bp = embT + j + (size_t)(2 * h) * NE; // K-half base
    v8f acc[4];
#pragma unroll
    for (int s = 0; s < 4; ++s) acc[s] = (v8f){};

#pragma unroll 4
    for (int kk = 0; kk < 256; kk += 4) {
      v2f bf;
      bf[0] = bp[(size_t)kk * NE];        // B[kk+2h][j]
      bf[1] = bp[(size_t)(kk + 1) * NE];  // B[kk+2h+1][j]
      const int ak = kk + 2 * h;
#pragma unroll
      for (int s = 0; s < 4; ++s) {
        const v2f a = *(const v2f*)(&As[(s * 16 + nn) * LDA + ak]);
        acc[s] = __builtin_amdgcn_wmma_f32_16x16x4_f32(
            false, a, false, bf, (short)0, acc[s], false, false);
      }
    }
    // Running per-lane best over this lane's code column j.
#pragma unroll
    for (int s = 0; s < 4; ++s)
#pragma unroll
      for (int r = 0; r < 8; ++r) {
        const float v = acc[s][r];
        const int q = s * 8 + r;
        if (v > bv[q]) { bv[q] = v; bi[q] = j; }   // strict > keeps smallest j
      }
  }

  // Argmax across the 16 code columns (16-lane halves), tie -> smaller index.
#pragma unroll
  for (int q = 0; q < 32; ++q) {
    float v = bv[q]; int i = bi[q];
#pragma unroll
    for (int m = 1; m < 16; m <<= 1) {
      const float ov = __shfl_xor(v, m, 16);
      const int   oi = __shfl_xor(i, m, 16);
      if (ov > v || (ov == v && oi < i)) { v = ov; i = oi; }
    }
    bv[q] = v; bi[q] = i;
  }

  // Lane nn==0 of each half holds the per-wave best for its 32 tokens.
  if (nn == 0) {
#pragma unroll
    for (int s = 0; s < 4; ++s)
#pragma unroll
      for (int r = 0; r < 8; ++r) {
        const int tok = s * 16 + h * 8 + r;
        wbv[w * 64 + tok] = bv[s * 8 + r];
        wbi[w * 64 + tok] = bi[s * 8 + r];
      }
  }
  __syncthreads();

  // Combine the 8 waves (each covered different code strips).
  if (tid < 64) {
    float v = wbv[tid]; int i = wbi[tid];
#pragma unroll
    for (int ww = 1; ww < 8; ++ww) {
      const float v2 = wbv[ww * 64 + tid];
      const int   i2 = wbi[ww * 64 + tid];
      if (v2 > v || (v2 == v && i2 < i)) { v = v2; i = i2; }
    }
    pbv[(size_t)blockIdx.y * TTOT + t0 + tid] = v;
    pbi[(size_t)blockIdx.y * TTOT + t0 + tid] = i;
  }
}

// ---------------------------------------------------------------------------
// 4) Combine the 4 code splits, emit indices / one-hot / counts / loss sum.
// ---------------------------------------------------------------------------
__global__ __launch_bounds__(256)
void k_finalize(const float* __restrict__ pbv, const int* __restrict__ pbi,
                int* __restrict__ fidx, float* __restrict__ out,
                int* __restrict__ counts, float* __restrict__ lossac) {
  __shared__ float red[256];
  const int t = blockIdx.x * 256 + threadIdx.x;
  float v = pbv[t]; int i = pbi[t];
#pragma unroll
  for (int y = 1; y < 4; ++y) {
    const float v2 = pbv[(size_t)y * TTOT + t];
    const int   i2 = pbi[(size_t)y * TTOT + t];
    if (v2 > v || (v2 == v && i2 < i)) { v = v2; i = i2; }
  }
  fidx[t] = i;
  out[OFF_IDX + t] = (float)i;
  out[OFF_MINENC + (size_t)t * NE + i] = 1.0f;
  atomicAdd(&counts[i], 1);
  // |z_q - z_n|^2 = 2 - 2*dot for unit-norm vectors.
  red[threadIdx.x] = 2.0f - 2.0f * v;
  __syncthreads();
  for (int s = 128; s > 0; s >>= 1) {
    if (threadIdx.x < s) red[threadIdx.x] += red[threadIdx.x + s];
    __syncthreads();
  }
  if (threadIdx.x == 0) atomicAdd(lossac, red[0]);
}

// ---------------------------------------------------------------------------
// 5) quant[b, c, n] = embT[c, idx[b, n]] (coalesced writes, L2-resident gather)
// ---------------------------------------------------------------------------
__global__ __launch_bounds__(256)
void k_quant(const float* __restrict__ embT, const int* __restrict__ fidx,
             float* __restrict__ out) {
  const int row = blockIdx.x;       // b*256 + c
  const int b = row >> 8;
  const int c = row & 255;
  const float* ecol = embT + (size_t)c * NE;
  const int* idx = fidx + b * NTOK;
  float* o = out + OFF_QUANT + (size_t)row * NTOK;
  for (int n = threadIdx.x; n < NTOK; n += 256) o[n] = ecol[idx[n]];
}

// ---------------------------------------------------------------------------
// 6) Scalars: vq/commit loss, codebook usage, perplexity. Single block.
// ---------------------------------------------------------------------------
__global__ __launch_bounds__(256)
void k_scalars(const int* __restrict__ counts, const float* __restrict__ lossac,
               float* __restrict__ out) {
  __shared__ float s_ent[256];
  __shared__ float s_used[256];
  float ent = 0.0f, used = 0.0f;
  for (int j = threadIdx.x; j < NE; j += 256) {
    const int c = counts[j];
    if (c > 0) {
      used += 1.0f;
      const float p = (float)c * (1.0f / (float)TTOT);
      ent += p * logf(p + 1e-10f);
    }
  }
  s_ent[threadIdx.x] = ent;
  s_used[threadIdx.x] = used;
  __syncthreads();
  for (int s = 128; s > 0; s >>= 1) {
    if (threadIdx.x < s) {
      s_ent[threadIdx.x]  += s_ent[threadIdx.x + s];
      s_used[threadIdx.x] += s_used[threadIdx.x + s];
    }
    __syncthreads();
  }
  if (threadIdx.x == 0) {
    const float vq = lossac[0] / ((float)TTOT * (float)EDIM);
    out[OFF_VQ]     = vq;
    out[OFF_COMMIT] = 0.25f * vq;
    out[OFF_USAGE]  = s_used[0] / (float)NE;
    out[OFF_PERP]   = expf(-s_ent[0]);
  }
}

// ---------------------------------------------------------------------------
extern "C" void kernel_launch(void* const* d_in, const int* in_sizes, int n_in,
                              void* d_out, int out_size, void* d_ws, size_t ws_size,
                              hipStream_t stream) {
  const float* z   = (const float*)d_in[0];   // [4, 256, 2048]
  const float* emb = (const float*)d_in[1];   // [16384, 256]
  float* out = (float*)d_out;
  char*  ws  = (char*)d_ws;

  // Workspace layout (~25.6 MB total)
  float* zn     = (float*)(ws);               //  8,388,608 B
  float* embT   = (float*)(ws + 8388608);     // 16,777,216 B
  float* invn   = (float*)(ws + 25165824);    //     65,536 B
  float* pbv    = (float*)(ws + 25231360);    //    131,072 B
  int*   pbi    = (int*)  (ws + 25362432);    //    131,072 B
  int*   fidx   = (int*)  (ws + 25493504);    //     32,768 B
  int*   counts = (int*)  (ws + 25526272);    //     65,536 B
  float* lossac = (float*)(ws + 25591808);    //        256 B

  // Zero the one-hot output region and the accumulators (graph-capture safe).
  hipMemsetAsync(out + OFF_MINENC, 0, (size_t)TTOT * NE * sizeof(float), stream);
  hipMemsetAsync(counts, 0, NE * sizeof(int) + 256, stream);

  k_norm_z     <<<TTOT / 8, 256, 0, stream>>>(z, zn);
  k_emb_norms  <<<NE / 8, 256, 0, stream>>>(emb, invn);
  k_transpose  <<<dim3(NE / 32, EDIM / 32), 256, 0, stream>>>(emb, invn, embT);
  k_gemm_argmax<<<dim3(TTOT / 64, 4), 256, 0, stream>>>(zn, embT, pbv, pbi);
  k_finalize   <<<TTOT / 256, 256, 0, stream>>>(pbv, pbi, fidx, out, counts, lossac);
  k_quant      <<<BATCH * EDIM, 256, 0, stream>>>(embT, fidx, out);
  k_scalars    <<<1, 256, 0, stream>>>(counts, lossac, out);
}